// DigitConvolutionalModel_7146825580673
// MI455X (gfx1250) — compile-verified
//
#include <hip/hip_runtime.h>

typedef float v2f __attribute__((ext_vector_type(2)));
typedef float v8f __attribute__((ext_vector_type(8)));

#define SPB 16        // samples per block
#define THREADS 256   // 8 wave32

__global__ __launch_bounds__(THREADS) void fused_digit_kernel(
    const float* __restrict__ x,       // [B,784]
    const float* __restrict__ conv_w,  // [3,3]
    const float* __restrict__ fc1_w,   // [128,676]
    const float* __restrict__ fc1_b,   // [128]
    const float* __restrict__ fc2_w,   // [10,128]
    const float* __restrict__ fc2_b,   // [10]
    float* __restrict__ out)           // [B,10]
{
    __shared__ float convLds[SPB * 676];   // 43264 B, stride 676 is conflict-free for A-frags
    __shared__ float hLds[SPB * 132];      // 8448 B, padded stride
    __shared__ float w2Lds[10 * 132];      // 5280 B, padded stride
    __shared__ float b2Lds[10];

    const int tid  = threadIdx.x;
    const int lane = tid & 31;
    const int wave = tid >> 5;
    const int blockSample0 = blockIdx.x * SPB;

    // ---- stage FC2 weights/bias into LDS (covered by first barrier)
    for (int i = tid; i < 1280; i += THREADS)
        w2Lds[(i >> 7) * 132 + (i & 127)] = fc2_w[i];
    if (tid < 10) b2Lds[tid] = fc2_b[tid];

    // ---- conv weights (uniform scalar loads)
    const float w00 = conv_w[0], w01 = conv_w[1], w02 = conv_w[2];
    const float w10 = conv_w[3], w11 = conv_w[4], w12 = conv_w[5];
    const float w20 = conv_w[6], w21 = conv_w[7], w22 = conv_w[8];

    // ---- conv: each wave does 2 samples; lane = output column (0..25)
    // sliding 3-row register window: 3 coalesced loads + 9 FMA per output
    const int c = lane;
    if (c < 26) {
        for (int i = 0; i < 2; ++i) {
            const int sLocal = wave * 2 + i;
            const float* xs = x + (size_t)(blockSample0 + sLocal) * 784;
            float r0a = xs[c],      r0b = xs[c + 1],      r0c = xs[c + 2];
            float r1a = xs[28 + c], r1b = xs[28 + c + 1], r1c = xs[28 + c + 2];
            float* cl = &convLds[sLocal * 676 + c];
            #pragma unroll
            for (int r = 0; r < 26; ++r) {
                const float* xr = xs + (r + 2) * 28 + c;
                float r2a = xr[0], r2b = xr[1], r2c = xr[2];
                float o = r0a * w00 + r0b * w01 + r0c * w02
                        + r1a * w10 + r1b * w11 + r1c * w12
                        + r2a * w20 + r2b * w21 + r2c * w22;
                cl[r * 26] = o;
                r0a = r1a; r0b = r1b; r0c = r1c;
                r1a = r2a; r1b = r2b; r1c = r2c;
            }
        }
    }
    __syncthreads();

    // ---- FC1 GEMM: each wave computes a 16(M=samples) x 16(N=wave*16..) tile,
    // K=676 via 169x V_WMMA_F32_16X16X4_F32 (full f32 precision).
    // A frag (16x4 f32): lanes 0-15 hold M=lane K={k4,k4+1}; lanes 16-31 K={k4+2,k4+3}
    // B frag (4x16 f32): lane holds B[koff..koff+1][n] = fc1_w[n][k4+koff..+1] (contig)
    const int nbase = wave * 16;
    const int l15  = lane & 15;
    const int koff = (lane >> 4) * 2;
    const float* ap = &convLds[l15 * 676 + koff];
    const float* bp = fc1_w + (size_t)(nbase + l15) * 676 + koff;

    v8f acc = {};
    #pragma unroll 4
    for (int k4 = 0; k4 < 676; k4 += 4) {
        v2f a = *(const v2f*)(ap + k4);
        v2f b = *(const v2f*)(bp + k4);
        acc = __builtin_amdgcn_wmma_f32_16x16x4_f32(
            /*neg_a=*/false, a, /*neg_b=*/false, b,
            /*c_mod=*/(short)0, acc, /*reuse_a=*/false, /*reuse_b=*/false);
    }

    // ---- epilogue: bias + ReLU -> hLds  (C/D layout: VGPR v -> M = v (+8 for hi half))
    const int nC    = nbase + l15;
    const int mBase = (lane >> 4) * 8;
    const float bias1 = fc1_b[nC];
    #pragma unroll
    for (int v = 0; v < 8; ++v) {
        float hv = acc[v] + bias1;
        hv = hv > 0.0f ? hv : 0.0f;
        hLds[(mBase + v) * 132 + nC] = hv;
    }
    __syncthreads();

    // ---- FC2: 16 samples x 10 outputs = 160 tasks, 128-MAC dot each (float4)
    if (tid < 160) {
        const int s = tid / 10, o = tid % 10;
        const float* hp = &hLds[s * 132];
        const float* wp = &w2Lds[o * 132];
        float sum = 0.0f;
        #pragma unroll 8
        for (int k = 0; k < 128; k += 4) {
            float4 hv = *(const float4*)(hp + k);
            float4 wv = *(const float4*)(wp + k);
            sum += hv.x * wv.x + hv.y * wv.y + hv.z * wv.z + hv.w * wv.w;
        }
        out[(size_t)(blockSample0 + s) * 10 + o] = sum + b2Lds[o];
    }
}

extern "C" void kernel_launch(void* const* d_in, const int* in_sizes, int n_in,
                              void* d_out, int out_size, void* d_ws, size_t ws_size,
                              hipStream_t stream) {
    const float* x      = (const float*)d_in[0];
    const float* conv_w = (const float*)d_in[1];
    const float* fc1_w  = (const float*)d_in[2];
    const float* fc1_b  = (const float*)d_in[3];
    const float* fc2_w  = (const float*)d_in[4];
    const float* fc2_b  = (const float*)d_in[5];
    float* out = (float*)d_out;

    const int B = in_sizes[0] / 784;   // 65536
    const int grid = B / SPB;          // 4096 blocks, 16 samples each
    fused_digit_kernel<<<grid, THREADS, 0, stream>>>(
        x, conv_w, fc1_w, fc1_b, fc2_w, fc2_b, out);
}